// DenseMap_36258113913067
// MI455X (gfx1250) — compile-verified
//
#include <hip/hip_runtime.h>
#include <hip/hip_bf16.h>

typedef __attribute__((ext_vector_type(2))) float v2f;
typedef __attribute__((ext_vector_type(8))) float v8f;

#define RES   128
#define FEAT  1024          // map_num(128) * feat_dim(8)
#define WAVES_PER_BLOCK 8   // wave32: 256-thread block = 8 waves = one point per wave

// One wave per point. Blend of 4 gathered feature rows done with
// V_WMMA_F32_16X16X4_F32:  D = A(16x4, weight rows replicated) x B(4x16 features).
// Every row of D is identical => D VGPR0 holds result for channel cbase+(lane&15)
// in all 32 lanes. Two channel-groups per loop iteration -> full-wave 128B store.
__global__ __launch_bounds__(256) void densemap_wmma_kernel(
    const float* __restrict__ inputs,   // (batch, 2)
    const float* __restrict__ emb,      // (RES*RES, FEAT)
    float* __restrict__ out,            // (batch, FEAT)
    int batch)
{
    const int wave = threadIdx.x >> 5;
    const int lane = threadIdx.x & 31;
    const int l15  = lane & 15;
    const int half = lane >> 4;                 // 0: lanes 0-15, 1: lanes 16-31
    const int p    = blockIdx.x * WAVES_PER_BLOCK + wave;
    if (p >= batch) return;                     // whole-wave uniform; active waves keep EXEC all-1s

    // ---- per-point scalar setup (uniform across the wave) ----
    const float u0 = inputs[2 * p + 0];
    const float u1 = inputs[2 * p + 1];
    const float x0 = u0 * (float)(RES - 1);
    const float x1 = u1 * (float)(RES - 1);
    const float fl0 = floorf(x0), fl1 = floorf(x1);
    const float f0 = x0 - fl0,   f1 = x1 - fl1;
    const int   i0 = (int)fl0,   i1 = (int)fl1;

    // neighbor n: bit0 -> dim0 (+1 if set), bit1 -> dim1; id = n0*RES + n1
    const int id0 = i0 * RES + i1;              // w = (1-f0)(1-f1)
    const int id1 = (i0 + 1) * RES + i1;        // w = f0(1-f1)
    const int id2 = i0 * RES + (i1 + 1);        // w = (1-f0)f1
    const int id3 = (i0 + 1) * RES + (i1 + 1);  // w = f0*f1
    const float w0 = (1.f - f0) * (1.f - f1);
    const float w1 = f0 * (1.f - f1);
    const float w2 = (1.f - f0) * f1;
    const float w3 = f0 * f1;

    const float* __restrict__ r0 = emb + (size_t)id0 * FEAT;
    const float* __restrict__ r1 = emb + (size_t)id1 * FEAT;
    const float* __restrict__ r2 = emb + (size_t)id2 * FEAT;
    const float* __restrict__ r3 = emb + (size_t)id3 * FEAT;

    // A operand (16x4 f32): lanes 0-15 -> {K=0,K=1} = {w0,w1}; lanes 16-31 -> {K=2,K=3} = {w2,w3}
    v2f a;
    a.x = half ? w2 : w0;
    a.y = half ? w3 : w1;

    // B operand row pointers per half-wave: VGPR0 holds rows K=0|K=2, VGPR1 rows K=1|K=3
    const float* __restrict__ rb0 = half ? r2 : r0;
    const float* __restrict__ rb1 = half ? r3 : r1;

    float* __restrict__ outp = out + (size_t)p * FEAT;

    const v8f czero = {};
#pragma unroll 4
    for (int g = 0; g < FEAT; g += 32) {
        // two 16-channel groups: [g, g+16) and [g+16, g+32)
        v2f bA, bB;
        bA.x = rb0[g + l15];        // B[K=0|2, n=l15], temporal (L2-resident table)
        bA.y = rb1[g + l15];        // B[K=1|3, n=l15]
        bB.x = rb0[g + 16 + l15];
        bB.y = rb1[g + 16 + l15];

        // D = A x B + 0   (8 args: neg_a, A, neg_b, B, c_mod, C, reuse_a, reuse_b)
        v8f dA = __builtin_amdgcn_wmma_f32_16x16x4_f32(
            false, a, false, bA, (short)0, czero, false, false);
        v8f dB = __builtin_amdgcn_wmma_f32_16x16x4_f32(
            false, a, false, bB, (short)0, czero, false, false);

        // D VGPR0 lane L = result for channel (group base + (L&15)); rows identical.
        // lanes 0-15 cover g+l15 (dA), lanes 16-31 cover g+16+l15 (dB)
        const float val = half ? dB[0] : dA[0];
        // streaming output: non-temporal so the 256MB store doesn't evict the
        // 64MB embedding table from the 192MB L2
        __builtin_nontemporal_store(val, outp + g + lane);
    }
}

extern "C" void kernel_launch(void* const* d_in, const int* in_sizes, int n_in,
                              void* d_out, int out_size, void* d_ws, size_t ws_size,
                              hipStream_t stream) {
    const float* inputs = (const float*)d_in[0];   // (batch, 2) f32
    const float* emb    = (const float*)d_in[1];   // (16384, 1024) f32
    float* out          = (float*)d_out;           // (batch, 1024) f32

    const int batch  = in_sizes[0] / 2;
    const int blocks = (batch + WAVES_PER_BLOCK - 1) / WAVES_PER_BLOCK;

    hipLaunchKernelGGL(densemap_wmma_kernel, dim3(blocks), dim3(256), 0, stream,
                       inputs, emb, out, batch);
}